// RelativePositioning_7791070675399
// MI455X (gfx1250) — compile-verified
//
#include <hip/hip_runtime.h>
#include <hip/hip_bf16.h>

// out[b*16+h, i, j] = (i>=j) ? e1[i-j, h] : e2[j-i-1, h];  duplicated over b=0..1
// Pure 512 MiB write-stream -> bandwidth bound (~22us @ 23.3 TB/s).

#define SEQ 2048
#define NH  16
#define TI  32                 // rows per block
#define TJ  1024               // cols per block
#define NSTAGE (TJ + TI - 1)   // diagonal-table entries needed per tile = 1055
#define NTHREADS 256

typedef float v4f __attribute__((ext_vector_type(4)));

typedef __attribute__((address_space(1))) int* gptr_i32;
typedef __attribute__((address_space(3))) int* lptr_i32;
typedef __attribute__((address_space(3))) float* lptr_f32;

__device__ __forceinline__ gptr_i32 as_global_i32(const float* p) {
    // AS(1) addresses are value-identical to generic addresses for global mem.
    return (gptr_i32)(size_t)p;
}
__device__ __forceinline__ lptr_i32 as_lds_i32(float* p) {
    // real addrspacecast generic->AS(3), then same-AS pointee bitcast
    return (lptr_i32)(lptr_f32)p;
}

__global__ __launch_bounds__(NTHREADS)
void RelativePositioning_7791070675399_kernel(const float* __restrict__ e1,
                                              const float* __restrict__ e2,
                                              float* __restrict__ out) {
    __shared__ float cbuf[NSTAGE];

    const int tid = threadIdx.x;
    const int j0  = blockIdx.x * TJ;
    const int i0  = blockIdx.y * TI;
    const int h   = blockIdx.z;

    // Stage c[d] for d in [dmin, dmin+NSTAGE): c[d] = d>=0 ? e1[d,h] : e2[-d-1,h]
    const int dmin = i0 - j0 - (TJ - 1);

#if __has_builtin(__builtin_amdgcn_global_load_async_to_lds_b32)
    // gfx1250 async gather: per-lane global address -> per-lane LDS address,
    // tracked with ASYNCcnt (no VGPR staging of the data).
    for (int k = tid; k < NSTAGE; k += NTHREADS) {
        const int d = dmin + k;
        const float* src = (d >= 0) ? (e1 + (size_t)d * NH + h)
                                    : (e2 + (size_t)(-d - 1) * NH + h);
        __builtin_amdgcn_global_load_async_to_lds_b32(
            as_global_i32(src), as_lds_i32(&cbuf[k]), /*offset=*/0, /*cpol=*/0);
    }
  #if __has_builtin(__builtin_amdgcn_s_wait_asynccnt)
    __builtin_amdgcn_s_wait_asynccnt(0);
  #else
    asm volatile("s_wait_asynccnt 0x0" ::: "memory");
  #endif
#else
    for (int k = tid; k < NSTAGE; k += NTHREADS) {
        const int d = dmin + k;
        cbuf[k] = (d >= 0) ? e1[(size_t)d * NH + h]
                           : e2[(size_t)(-d - 1) * NH + h];
    }
#endif
    __syncthreads();

    // Each thread owns columns [4*tid, 4*tid+3] of the tile, all TI rows.
    float* base0 = out + (size_t)h * SEQ * SEQ + (size_t)i0 * SEQ + j0;
    float* base1 = base0 + (size_t)NH * SEQ * SEQ;   // batch duplicate (b=1)

    const int c0 = 4 * tid;
#pragma unroll 4
    for (int r = 0; r < TI; ++r) {
        // column j0+c maps to table index k = (TJ-1) + r - c  (reversed slice)
        const int kx = (TJ - 1) + r - c0;
        v4f v;
        v.x = cbuf[kx];
        v.y = cbuf[kx - 1];
        v.z = cbuf[kx - 2];
        v.w = cbuf[kx - 3];
        // 512 MiB streaming output, never re-read: non-temporal B128 stores.
        __builtin_nontemporal_store(v, (v4f*)(base0 + (size_t)r * SEQ + c0));
        __builtin_nontemporal_store(v, (v4f*)(base1 + (size_t)r * SEQ + c0));
    }
}

extern "C" void kernel_launch(void* const* d_in, const int* in_sizes, int n_in,
                              void* d_out, int out_size, void* d_ws, size_t ws_size,
                              hipStream_t stream) {
    (void)in_sizes; (void)n_in; (void)out_size; (void)d_ws; (void)ws_size;
    // d_in[0] = q (unused by the reference output), d_in[1] = e1, d_in[2] = e2
    const float* e1 = (const float*)d_in[1];
    const float* e2 = (const float*)d_in[2];
    float* out = (float*)d_out;

    dim3 grid(SEQ / TJ, SEQ / TI, NH);   // (2, 64, 16) = 2048 blocks
    RelativePositioning_7791070675399_kernel<<<grid, NTHREADS, 0, stream>>>(e1, e2, out);
}